// DebugGCN_19275813224660
// MI455X (gfx1250) — compile-verified
//
#include <hip/hip_runtime.h>
#include <hip/hip_bf16.h>
#include <math.h>

typedef __attribute__((ext_vector_type(2))) float v2f;
typedef __attribute__((ext_vector_type(8))) float v8f;

#define FIN   128
#define FHID  128
#define FOUT  40
#define FOUT_PAD 48

// ---------------- degree / normalization ----------------

__global__ void k_deg_init(float* deg, int n) {
    int i = blockIdx.x * blockDim.x + threadIdx.x;
    if (i < n) deg[i] = 1.0f;                      // self-loop
}

__global__ void k_deg_count(const int* __restrict__ dst, float* deg, int e) {
    int i = blockIdx.x * blockDim.x + threadIdx.x;
    if (i < e) atomicAdd(&deg[dst[i]], 1.0f);
}

__global__ void k_deg_rsqrt(float* deg, int n) {
    int i = blockIdx.x * blockDim.x + threadIdx.x;
    if (i < n) deg[i] = rsqrtf(deg[i]);            // deg >= 1 always
}

// ---------------- WMMA GEMM 1: xw1[N,128] = X[N,128] @ W1[128,128] ----------------
// One wave -> one 16-row M tile, 8 N-accumulators (full 128 cols).
// A fragment loaded once per k-step and reused across all 8 WMMAs.
// f32 A 16x4 frag: lane half h, lane l: v0 = A[l][k+2h], v1 = A[l][k+2h+1]
// f32 B 4x16 frag: v0 = B[k+2h][n0+l], v1 = B[k+2h+1][n0+l]
// C/D 16x16:       c[j] = C[m0 + j + 8h][n0 + l]

__global__ void k_gemm1(const float* __restrict__ X, const float* __restrict__ W,
                        float* __restrict__ XW, int n) {
    const int wave = threadIdx.x >> 5;             // 0..7 -> M sub-tile within block
    const int lane = threadIdx.x & 31;
    const int l = lane & 15;
    const int h = lane >> 4;
    const int m0 = (blockIdx.x * 8 + wave) << 4;
    if (m0 >= n) return;                           // whole-wave exit

    int arow = m0 + l; if (arow >= n) arow = n - 1;
    const float* __restrict__ A = X + (size_t)arow * FIN;

    v8f c[8] = {};
    for (int k = 0; k < FIN; k += 4) {
        v2f a;
        a[0] = A[k + 2 * h];
        a[1] = A[k + 2 * h + 1];
        const float* __restrict__ Wr0 = W + (size_t)(k + 2 * h) * FHID + l;
        const float* __restrict__ Wr1 = Wr0 + FHID;
        #pragma unroll
        for (int t = 0; t < 8; ++t) {
            v2f b;
            b[0] = Wr0[t << 4];
            b[1] = Wr1[t << 4];
            c[t] = __builtin_amdgcn_wmma_f32_16x16x4_f32(false, a, false, b,
                                                         (short)0, c[t], false, false);
        }
    }

    if (m0 + 16 <= n) {                            // full tile: unguarded stores
        #pragma unroll
        for (int j = 0; j < 8; ++j) {
            float* __restrict__ R = XW + (size_t)(m0 + j + 8 * h) * FHID + l;
            #pragma unroll
            for (int t = 0; t < 8; ++t) R[t << 4] = c[j][t];
        }
    } else {
        #pragma unroll
        for (int j = 0; j < 8; ++j) {
            int row = m0 + j + 8 * h;
            if (row < n) {
                float* __restrict__ R = XW + (size_t)row * FHID + l;
                #pragma unroll
                for (int t = 0; t < 8; ++t) R[t << 4] = c[j][t];
            }
        }
    }
}

// ---------------- WMMA GEMM 2: xw2[N,48] = H[N,128] @ W2[128,40] (N-padded) ------

__global__ void k_gemm2(const float* __restrict__ H, const float* __restrict__ W,
                        float* __restrict__ XW, int n) {
    const int wave = threadIdx.x >> 5;
    const int lane = threadIdx.x & 31;
    const int l = lane & 15;
    const int h = lane >> 4;
    const int m0 = (blockIdx.x * 8 + wave) << 4;
    if (m0 >= n) return;

    int arow = m0 + l; if (arow >= n) arow = n - 1;
    const float* __restrict__ A = H + (size_t)arow * FHID;
    const bool tail_ok = (32 + l) < FOUT;          // mask for N-tile 2 (cols 32..47)

    v8f c[3] = {};
    for (int k = 0; k < FHID; k += 4) {
        v2f a;
        a[0] = A[k + 2 * h];
        a[1] = A[k + 2 * h + 1];
        const float* __restrict__ Wr0 = W + (size_t)(k + 2 * h) * FOUT + l;
        const float* __restrict__ Wr1 = Wr0 + FOUT;
        v2f b0, b1, b2;
        b0[0] = Wr0[0];  b0[1] = Wr1[0];
        b1[0] = Wr0[16]; b1[1] = Wr1[16];
        b2[0] = tail_ok ? Wr0[32] : 0.0f;
        b2[1] = tail_ok ? Wr1[32] : 0.0f;
        c[0] = __builtin_amdgcn_wmma_f32_16x16x4_f32(false, a, false, b0,
                                                     (short)0, c[0], false, false);
        c[1] = __builtin_amdgcn_wmma_f32_16x16x4_f32(false, a, false, b1,
                                                     (short)0, c[1], false, false);
        c[2] = __builtin_amdgcn_wmma_f32_16x16x4_f32(false, a, false, b2,
                                                     (short)0, c[2], false, false);
    }

    if (m0 + 16 <= n) {
        #pragma unroll
        for (int j = 0; j < 8; ++j) {
            float* __restrict__ R = XW + (size_t)(m0 + j + 8 * h) * FOUT_PAD + l;
            R[0] = c[0][j]; R[16] = c[1][j]; R[32] = c[2][j];
        }
    } else {
        #pragma unroll
        for (int j = 0; j < 8; ++j) {
            int row = m0 + j + 8 * h;
            if (row < n) {
                float* __restrict__ R = XW + (size_t)row * FOUT_PAD + l;
                R[0] = c[0][j]; R[16] = c[1][j]; R[32] = c[2][j];
            }
        }
    }
}

// ---------------- layer-1 aggregation ----------------

__global__ void k_selfloop1(const float* __restrict__ xw, const float* __restrict__ dinv,
                            float* __restrict__ h, int n) {
    int t = blockIdx.x * blockDim.x + threadIdx.x;           // n*32 threads, 4 feats each
    if (t >= n * (FHID / 4)) return;
    int i = t >> 5;
    int f = (t & 31) << 2;
    float s = dinv[i]; s = s * s;
    const float4 v = *(const float4*)(xw + (size_t)i * FHID + f);
    float4 o; o.x = v.x * s; o.y = v.y * s; o.z = v.z * s; o.w = v.w * s;
    *(float4*)(h + (size_t)i * FHID + f) = o;
}

__global__ void k_edge_agg1(const int* __restrict__ src, const int* __restrict__ dst,
                            const float* __restrict__ dinv, const float* __restrict__ xw,
                            float* __restrict__ h, int e) {
    int t = blockIdx.x * blockDim.x + threadIdx.x;           // e*32 threads, 4 feats each
    if (t >= e * (FHID / 4)) return;
    int ed = t >> 5;
    int f  = (t & 31) << 2;
    int s = src[ed], d = dst[ed];
    float norm = dinv[s] * dinv[d];
    const float4 v = *(const float4*)(xw + (size_t)s * FHID + f);
    float* out = h + (size_t)d * FHID + f;
    atomicAdd(out + 0, v.x * norm);
    atomicAdd(out + 1, v.y * norm);
    atomicAdd(out + 2, v.z * norm);
    atomicAdd(out + 3, v.w * norm);
}

__global__ void k_bias_relu(float* __restrict__ h, const float* __restrict__ b, int n) {
    int t = blockIdx.x * blockDim.x + threadIdx.x;
    if (t >= n * (FHID / 4)) return;
    int i = t >> 5;
    int f = (t & 31) << 2;
    float4 v = *(float4*)(h + (size_t)i * FHID + f);
    const float4 bb = *(const float4*)(b + f);
    v.x = fmaxf(v.x + bb.x, 0.0f);
    v.y = fmaxf(v.y + bb.y, 0.0f);
    v.z = fmaxf(v.z + bb.z, 0.0f);
    v.w = fmaxf(v.w + bb.w, 0.0f);
    *(float4*)(h + (size_t)i * FHID + f) = v;
}

// ---------------- layer-2 aggregation (into d_out, width 40) ----------------

__global__ void k_selfloop2(const float* __restrict__ xw2, const float* __restrict__ dinv,
                            float* __restrict__ out, int n) {
    int t = blockIdx.x * blockDim.x + threadIdx.x;           // n*10 threads, 4 feats each
    if (t >= n * (FOUT / 4)) return;
    int i = t / (FOUT / 4);
    int f = (t % (FOUT / 4)) << 2;
    float s = dinv[i]; s = s * s;
    const float4 v = *(const float4*)(xw2 + (size_t)i * FOUT_PAD + f);
    float4 o; o.x = v.x * s; o.y = v.y * s; o.z = v.z * s; o.w = v.w * s;
    *(float4*)(out + (size_t)i * FOUT + f) = o;
}

__global__ void k_edge_agg2(const int* __restrict__ src, const int* __restrict__ dst,
                            const float* __restrict__ dinv, const float* __restrict__ xw2,
                            float* __restrict__ out, int e) {
    int t = blockIdx.x * blockDim.x + threadIdx.x;           // e*10 threads, 4 feats each
    if (t >= e * (FOUT / 4)) return;
    int ed = t / (FOUT / 4);
    int f  = (t % (FOUT / 4)) << 2;
    int s = src[ed], d = dst[ed];
    float norm = dinv[s] * dinv[d];
    const float4 v = *(const float4*)(xw2 + (size_t)s * FOUT_PAD + f);
    float* o = out + (size_t)d * FOUT + f;
    atomicAdd(o + 0, v.x * norm);
    atomicAdd(o + 1, v.y * norm);
    atomicAdd(o + 2, v.z * norm);
    atomicAdd(o + 3, v.w * norm);
}

// ---------------- bias + log_softmax (in place on d_out) ----------------

__global__ void k_logsoftmax(float* __restrict__ out, const float* __restrict__ b2, int n) {
    int i = blockIdx.x * blockDim.x + threadIdx.x;
    if (i >= n) return;
    float* row = out + (size_t)i * FOUT;
    float v[FOUT];
    float m = -INFINITY;
    #pragma unroll
    for (int f = 0; f < FOUT; ++f) { v[f] = row[f] + b2[f]; m = fmaxf(m, v[f]); }
    float ssum = 0.0f;
    #pragma unroll
    for (int f = 0; f < FOUT; ++f) ssum += __expf(v[f] - m);
    float lse = m + __logf(ssum);
    #pragma unroll
    for (int f = 0; f < FOUT; ++f) row[f] = v[f] - lse;
}

// ---------------- host launcher ----------------

extern "C" void kernel_launch(void* const* d_in, const int* in_sizes, int n_in,
                              void* d_out, int out_size, void* d_ws, size_t ws_size,
                              hipStream_t stream) {
    const float* x  = (const float*)d_in[0];
    const int*   ei = (const int*)d_in[1];        // [2, E] flat: src then dst
    const float* W1 = (const float*)d_in[2];
    const float* b1 = (const float*)d_in[3];
    const float* W2 = (const float*)d_in[4];
    const float* b2 = (const float*)d_in[5];
    float* out = (float*)d_out;

    const int N = in_sizes[0] / FIN;              // 100000
    const int E = in_sizes[1] / 2;                // 640000
    const int* src = ei;
    const int* dst = ei + E;

    // workspace carve-up (floats)
    float* dinv = (float*)d_ws;                                   // N
    float* xw1  = dinv + N;                                       // N*128
    float* h1   = xw1 + (size_t)N * FHID;                         // N*128
    float* xw2  = h1 + (size_t)N * FHID;                          // N*48
    (void)ws_size; (void)n_in; (void)out_size;

    const int B = 256;

    // 1) normalization coefficients
    k_deg_init <<<(N + B - 1) / B, B, 0, stream>>>(dinv, N);
    k_deg_count<<<(E + B - 1) / B, B, 0, stream>>>(dst, dinv, E);
    k_deg_rsqrt<<<(N + B - 1) / B, B, 0, stream>>>(dinv, N);

    // 2) layer 1: GEMM (WMMA fp32) -> self-loop seed -> edge scatter -> bias+ReLU
    const int mtiles = (N + 15) / 16;             // 6250
    const int gblocks = (mtiles + 7) / 8;         // 8 waves/block, 1 M-tile/wave
    k_gemm1<<<gblocks, 256, 0, stream>>>(x, W1, xw1, N);

    int t1 = N * (FHID / 4);
    k_selfloop1<<<(t1 + B - 1) / B, B, 0, stream>>>(xw1, dinv, h1, N);
    int te1 = E * (FHID / 4);
    k_edge_agg1<<<(te1 + B - 1) / B, B, 0, stream>>>(src, dst, dinv, xw1, h1, E);
    k_bias_relu<<<(t1 + B - 1) / B, B, 0, stream>>>(h1, b1, N);

    // 3) layer 2: GEMM (WMMA fp32, N padded to 48) -> self-loop -> edge scatter
    k_gemm2<<<gblocks, 256, 0, stream>>>(h1, W2, xw2, N);

    int t2 = N * (FOUT / 4);
    k_selfloop2<<<(t2 + B - 1) / B, B, 0, stream>>>(xw2, dinv, out, N);
    int te2 = E * (FOUT / 4);
    k_edge_agg2<<<(te2 + B - 1) / B, B, 0, stream>>>(src, dst, dinv, xw2, out, E);

    // 4) bias + log_softmax in place
    k_logsoftmax<<<(N + B - 1) / B, B, 0, stream>>>(out, b2, N);
}